// GroupedGRUStep_55765855371909
// MI455X (gfx1250) — compile-verified
//
#include <hip/hip_runtime.h>
#include <hip/hip_bf16.h>

typedef __attribute__((ext_vector_type(2))) float v2f;
typedef __attribute__((ext_vector_type(8))) float v8f;

#define GRU_G     8
#define IPG       64
#define HPG       64
#define NCOL      192   // 3*HPG
#define LDS_PCOL  208   // padded pair-row length: +1 pair-row == +416 dwords == +32 banks,
                        // so half-wave 0 uses banks 0..31 and half-wave 1 uses 32..63
#define UOFF      (32 * LDS_PCOL * 2)  // float offset of U matrix in combined LDS array
#define HSTRIDE   1536  // L*H
#define XSTRIDE   512   // H (and I)

// LDS layout: element [k][c] of a 64x192 weight matrix lives at
//   ((k>>1)*LDS_PCOL + c)*2 + (k&1)
// so the (K, K+1) pair needed by one WMMA B-fragment is a single aligned b64.
// Per-matrix chain offsets span < 53KB -> fit the 16-bit DS immediate, provided
// each matrix keeps its own base register (enforced via the asm launder below).

__device__ __forceinline__ v8f mm_chain(v8f c, const v2f a[16],
                                        const float* base, int colbase) {
#pragma unroll
  for (int kk = 0; kk < 16; ++kk) {
    const v2f bf = *(const v2f*)(base + kk * (4 * LDS_PCOL) + colbase * 2);
    c = __builtin_amdgcn_wmma_f32_16x16x4_f32(false, a[kk], false, bf,
                                              (short)0, c, false, false);
  }
  return c;
}

__global__ void __launch_bounds__(256)
gru_layer_kernel(const float* __restrict__ x,      // (B, 512)
                 const float* __restrict__ hin,    // (B, 1536) original states
                 const float* __restrict__ Wl,     // (G, 64, 192) this layer
                 const float* __restrict__ Ul,     // (G, 64, 192)
                 const float* __restrict__ bl,     // (G, 2, 192)
                 float* __restrict__ hout,         // (B, 1536) new_states
                 float* __restrict__ xnext,        // (B, 512) next-layer x / output
                 int loff,                         // l * 512
                 int do_shuffle) {
  __shared__ float ldsWU[2 * UOFF];                // 106,496 B total (W then U)

  const int g   = blockIdx.y;
  const int tid = threadIdx.x;

  // Stage this group's W and U into LDS, K-pair interleaved.
  const float* Wg = Wl + (size_t)g * IPG * NCOL;
  const float* Ug = Ul + (size_t)g * HPG * NCOL;
  for (int u = tid; u < 32 * NCOL; u += 256) {
    const int pk = u / NCOL, c = u % NCOL;
    v2f w, uu;
    w.x  = Wg[(2 * pk) * NCOL + c];
    w.y  = Wg[(2 * pk + 1) * NCOL + c];
    uu.x = Ug[(2 * pk) * NCOL + c];
    uu.y = Ug[(2 * pk + 1) * NCOL + c];
    *(v2f*)(ldsWU + ((pk * LDS_PCOL + c) << 1))        = w;
    *(v2f*)(ldsWU + UOFF + ((pk * LDS_PCOL + c) << 1)) = uu;
  }
  __syncthreads();

  const int wave = tid >> 5;
  const int lane = tid & 31;
  const int half = lane >> 4;
  const int n    = lane & 15;           // A-row / B-col / C-col lane index
  const int r0   = (blockIdx.x * 8 + wave) * 16;

  // A fragments: lane holds row M=n; VGPR pair = K = 4*kk + 2*half {+0,+1}.
  // 32-bit element offsets (max ~25M elements, fits int), widened once.
  v2f ax[16], ah[16];
  const float* xrow = x   + (size_t)((r0 + n) * XSTRIDE + g * IPG + 2 * half);
  const float* hrow = hin + (size_t)((r0 + n) * HSTRIDE + loff + g * HPG + 2 * half);
#pragma unroll
  for (int kk = 0; kk < 16; ++kk) {
    ax[kk] = *(const v2f*)(xrow + 4 * kk);
    ah[kk] = *(const v2f*)(hrow + 4 * kk);
  }

  // Per-lane LDS fragment bases. Launder the offsets so LLVM keeps two
  // separate base registers and emits immediate-offset ds_load_b64 for
  // every fragment (each matrix's span fits the 16-bit DS offset).
  int woff = 2 * n + half * (2 * LDS_PCOL);
  int uoff = woff + UOFF;
  asm volatile("" : "+v"(woff), "+v"(uoff));
  const float* wfb = ldsWU + woff;
  const float* ufb = ldsWU + uoff;

  const float* bWg = bl + g * 2 * NCOL;   // bias[g][0][:] (input bias)
  const float* bUg = bWg + NCOL;          // bias[g][1][:] (recurrent bias)

  // Per-lane bases for C-layout global accesses (row M = v + 8*half).
  const int cbase = (r0 + 8 * half) * HSTRIDE + loff + g * HPG + n;
  const float* hold_b = hin  + (size_t)cbase;
  float*       hout_b = hout + (size_t)cbase;
  // Shuffle p = (f&7)*64 + (f>>3) with f = g*64 + 16t + n collapses to pbase + 2t.
  const int colb  = do_shuffle ? ((n & 7) * 64 + g * 8 + (n >> 3)) : (g * HPG + n);
  const int cstep = do_shuffle ? 2 : 16;
  float*    xn_b  = xnext + (size_t)((r0 + 8 * half) * XSTRIDE + colb);

#pragma unroll
  for (int t = 0; t < 4; ++t) {           // 4 output tiles of 16 cols (HPG=64)
    const int cz = t * 16, cr = 64 + t * 16, ch = 128 + t * 16;

    // Zero-init C -> first WMMA uses inline 0 SRC2; bias folded into epilogue.
    v8f gz = {}, gr = {}, gxh = {}, ghh = {};
    gz  = mm_chain(gz,  ax, wfb, cz);     // x*Wz
    gz  = mm_chain(gz,  ah, ufb, cz);     // + h*Uz   (z gate fused)
    gr  = mm_chain(gr,  ax, wfb, cr);
    gr  = mm_chain(gr,  ah, ufb, cr);     // r gate fused
    gxh = mm_chain(gxh, ax, wfb, ch);     // candidate stays split:
    ghh = mm_chain(ghh, ah, ufb, ch);     // hcand = tanh(xh + r*hh)

    const float bz  = bWg[cz + n] + bUg[cz + n];
    const float br  = bWg[cr + n] + bUg[cr + n];
    const float bxh = bWg[ch + n];
    const float bhh = bUg[ch + n];
    float* xnt = xn_b + t * cstep;

#pragma unroll
    for (int v = 0; v < 8; ++v) {
      const float hold = hold_b[v * HSTRIDE + t * 16];
      const float z    = 1.f / (1.f + __expf(-(gz[v] + bz)));
      const float rr   = 1.f / (1.f + __expf(-(gr[v] + br)));
      const float hc   = tanhf(gxh[v] + bxh + rr * (ghh[v] + bhh));
      const float o    = z * hold + (1.f - z) * hc;
      hout_b[v * HSTRIDE + t * 16] = o;
      xnt[v * XSTRIDE] = o;
    }
  }
}

extern "C" void kernel_launch(void* const* d_in, const int* in_sizes, int n_in,
                              void* d_out, int out_size, void* d_ws, size_t ws_size,
                              hipStream_t stream) {
  const float* x_t    = (const float*)d_in[0];   // (B, 512)
  const float* states = (const float*)d_in[1];   // (B, 1536)
  const float* W      = (const float*)d_in[2];   // (3, 8, 64, 192)
  const float* U      = (const float*)d_in[3];   // (3, 8, 64, 192)
  const float* bias   = (const float*)d_in[4];   // (3, 8, 2, 192)

  const int B = in_sizes[0] / XSTRIDE;           // 16384
  float* out        = (float*)d_out;             // [0, B*512): output
  float* new_states = out + (size_t)B * XSTRIDE; // [B*512, B*512 + B*1536)
  float* xbuf0      = out;                       // layer0 shuffled x (overwritten by layer2)
  float* xbuf1      = (float*)d_ws;              // layer1 shuffled x (B*512 floats)

  const size_t wsz = (size_t)GRU_G * IPG * NCOL; // per-layer W/U stride
  const size_t bsz = (size_t)GRU_G * 2 * NCOL;   // per-layer bias stride

  dim3 grid(B / 128, GRU_G), block(256);
  // Layer 0: x_t -> h0; shuffled x into out-region scratch
  gru_layer_kernel<<<grid, block, 0, stream>>>(x_t, states, W, U, bias,
                                               new_states, xbuf0, 0, 1);
  // Layer 1: xbuf0 -> h1; shuffled x into ws scratch
  gru_layer_kernel<<<grid, block, 0, stream>>>(xbuf0, states, W + wsz, U + wsz,
                                               bias + bsz, new_states, xbuf1,
                                               XSTRIDE, 1);
  // Layer 2: xbuf1 -> h2; unshuffled result doubles as the final output
  gru_layer_kernel<<<grid, block, 0, stream>>>(xbuf1, states, W + 2 * wsz,
                                               U + 2 * wsz, bias + 2 * bsz,
                                               new_states, out, 2 * XSTRIDE, 0);
}